// DemandGNN_27367531610530
// MI455X (gfx1250) — compile-verified
//
#include <hip/hip_runtime.h>
#include <hip/hip_bf16.h>
#include <stdint.h>

typedef __attribute__((ext_vector_type(2))) float v2f;
typedef __attribute__((ext_vector_type(8))) float v8f;

#define HID 32
#define FEAT_IN 128

// ---------------- degree / normalization ----------------

__global__ void init_deg_kernel(float* deg, int n) {
  int i = blockIdx.x * blockDim.x + threadIdx.x;
  if (i < n) deg[i] = 1.0f;  // self-loop contributes 1 to every node's degree
}

__global__ void accum_deg_kernel(const long long* __restrict__ dst, float* deg, int E) {
  int e = blockIdx.x * blockDim.x + threadIdx.x;
  if (e < E) atomicAdd(&deg[(int)dst[e]], 1.0f);
}

__global__ void dinv_kernel(float* deg, int n) {
  int i = blockIdx.x * blockDim.x + threadIdx.x;
  if (i < n) deg[i] = rsqrtf(deg[i]);  // deg >= 1 always (self loop)
}

// ---------------- dense transform: out[N x 32] = A[N x K] @ W[K x 32] ----------------
// One wave per 16x16 output tile. blockDim = 64 (2 waves): wave w -> columns [16w,16w+16).
// V_WMMA_F32_16X16X4_F32 operand layouts (ISA 7.12.2):
//   A 16x4: lanes 0-15 row M=lane  {v0:K=k0,   v1:K=k0+1}
//           lanes16-31 row M=lane-16 {v0:K=k0+2, v1:K=k0+3}
//   B 4x16: lanes 0-15 col N=lane  {v0:K=k0,   v1:K=k0+1}
//           lanes16-31 col N=lane-16 {v0:K=k0+2, v1:K=k0+3}
//   D 16x16: VGPR r -> (M=r, N=lane) for lanes 0-15, (M=r+8, N=lane-16) for lanes 16-31
template <int K>
__global__ void gemm_wmma_kernel(const float* __restrict__ A,
                                 const float* __restrict__ W,
                                 float* __restrict__ out) {
  const int lane    = threadIdx.x & 31;
  const int wave    = threadIdx.x >> 5;   // 0/1 -> column tile
  const int rowBase = blockIdx.x << 4;    // 16 rows per block
  const int colBase = wave << 4;
  const int half    = lane >> 4;          // which K-pair this lane carries
  const int m       = lane & 15;

  const float* arow = A + (size_t)(rowBase + m) * K;
  v8f acc = {};
#pragma unroll
  for (int k0 = 0; k0 < K; k0 += 4) {
    v2f a, b;
    a.x = arow[k0 + 2 * half];
    a.y = arow[k0 + 2 * half + 1];
    b.x = W[(size_t)(k0 + 2 * half) * HID + colBase + m];
    b.y = W[(size_t)(k0 + 2 * half + 1) * HID + colBase + m];
    acc = __builtin_amdgcn_wmma_f32_16x16x4_f32(false, a, false, b,
                                                (short)0, acc, false, false);
  }
  float* ocol = out + colBase + m;
#pragma unroll
  for (int r = 0; r < 8; ++r)
    ocol[(size_t)(rowBase + r + 8 * half) * HID] = acc[r];
}

// ---------------- aggregation ----------------

// agg = h * dinv^2  (self-loop message), elementwise over N*32
__global__ void selfloop_init_kernel(const float* __restrict__ h,
                                     const float* __restrict__ dinv,
                                     float* __restrict__ agg, int total) {
  int i = blockIdx.x * blockDim.x + threadIdx.x;
  if (i < total) {
    float w = dinv[i >> 5];
    agg[i] = h[i] * w * w;
  }
}

// one wave per edge, one feature per lane: agg[dst] += h[src] * dinv[src]*dinv[dst]
__global__ void scatter32_kernel(const long long* __restrict__ src,
                                 const long long* __restrict__ dst,
                                 const float* __restrict__ h,
                                 const float* __restrict__ dinv,
                                 float* __restrict__ agg, int E) {
  int e = blockIdx.x * (blockDim.x >> 5) + (threadIdx.x >> 5);
  if (e >= E) return;
  int lane = threadIdx.x & 31;
  int s = (int)src[e];
  int d = (int)dst[e];
  float w = dinv[s] * dinv[d];
  atomicAdd(&agg[(size_t)d * HID + lane], h[(size_t)s * HID + lane] * w);
}

// h = relu(agg + b)
__global__ void bias_relu_kernel(const float* __restrict__ agg,
                                 const float* __restrict__ b,
                                 float* __restrict__ h, int total) {
  int i = blockIdx.x * blockDim.x + threadIdx.x;
  if (i < total) {
    float v = agg[i] + b[i & 31];
    h[i] = v > 0.0f ? v : 0.0f;
  }
}

// ---------------- layer 3 (out dim = 1) ----------------

__global__ void dot_w3_kernel(const float* __restrict__ h,
                              const float* __restrict__ W3,
                              float* __restrict__ t, int n) {
  int i = blockIdx.x * blockDim.x + threadIdx.x;
  if (i < n) {
    const float* row = h + (size_t)i * HID;
    float s = 0.0f;
#pragma unroll
    for (int k = 0; k < HID; ++k) s += row[k] * W3[k];
    t[i] = s;
  }
}

__global__ void out_init_kernel(const float* __restrict__ t,
                                const float* __restrict__ dinv,
                                const float* __restrict__ b3,
                                float* __restrict__ out, int n) {
  int i = blockIdx.x * blockDim.x + threadIdx.x;
  if (i < n) {
    float w = dinv[i];
    out[i] = t[i] * w * w + b3[0];
  }
}

__global__ void scatter1_kernel(const long long* __restrict__ src,
                                const long long* __restrict__ dst,
                                const float* __restrict__ t,
                                const float* __restrict__ dinv,
                                float* __restrict__ out, int E) {
  int e = blockIdx.x * blockDim.x + threadIdx.x;
  if (e < E) {
    int s = (int)src[e];
    int d = (int)dst[e];
    atomicAdd(&out[d], t[s] * dinv[s] * dinv[d]);
  }
}

// ---------------- launcher ----------------

extern "C" void kernel_launch(void* const* d_in, const int* in_sizes, int n_in,
                              void* d_out, int out_size, void* d_ws, size_t ws_size,
                              hipStream_t stream) {
  const float*     x   = (const float*)d_in[0];
  const long long* ei  = (const long long*)d_in[1];  // int64 (2, E)
  const float*     W1  = (const float*)d_in[2];
  const float*     b1  = (const float*)d_in[3];
  const float*     W2  = (const float*)d_in[4];
  const float*     b2  = (const float*)d_in[5];
  const float*     W3  = (const float*)d_in[6];
  const float*     b3  = (const float*)d_in[7];
  float* out = (float*)d_out;

  const int N = in_sizes[0] / FEAT_IN;   // 100000 (divisible by 16)
  const int E = in_sizes[1] / 2;         // 3200000
  const long long* src = ei;
  const long long* dst = ei + E;

  // workspace layout (floats): dinv[N] | bufA[N*32] | bufB[N*32] | t[N]
  float* ws   = (float*)d_ws;
  size_t nPad = ((size_t)N + 255) & ~(size_t)255;
  float* dinv = ws;
  float* bufA = dinv + nPad;
  float* bufB = bufA + (size_t)N * HID;
  float* tvec = bufB + (size_t)N * HID;

  const int total = N * HID;
  const int TB = 256;

  // normalization coefficients
  init_deg_kernel<<<(N + TB - 1) / TB, TB, 0, stream>>>(dinv, N);
  accum_deg_kernel<<<(E + TB - 1) / TB, TB, 0, stream>>>(dst, dinv, E);
  dinv_kernel<<<(N + TB - 1) / TB, TB, 0, stream>>>(dinv, N);

  // ---- layer 1: x[N,128] @ W1 -> bufA; aggregate -> bufB; relu -> bufA
  gemm_wmma_kernel<FEAT_IN><<<N / 16, 64, 0, stream>>>(x, W1, bufA);
  selfloop_init_kernel<<<(total + TB - 1) / TB, TB, 0, stream>>>(bufA, dinv, bufB, total);
  scatter32_kernel<<<(E + 7) / 8, TB, 0, stream>>>(src, dst, bufA, dinv, bufB, E);
  bias_relu_kernel<<<(total + TB - 1) / TB, TB, 0, stream>>>(bufB, b1, bufA, total);

  // ---- layer 2: bufA[N,32] @ W2 -> bufB; aggregate -> bufA; relu -> bufB
  gemm_wmma_kernel<HID><<<N / 16, 64, 0, stream>>>(bufA, W2, bufB);
  selfloop_init_kernel<<<(total + TB - 1) / TB, TB, 0, stream>>>(bufB, dinv, bufA, total);
  scatter32_kernel<<<(E + 7) / 8, TB, 0, stream>>>(src, dst, bufB, dinv, bufA, E);
  bias_relu_kernel<<<(total + TB - 1) / TB, TB, 0, stream>>>(bufA, b2, bufB, total);

  // ---- layer 3: dot with W3[32,1]; aggregate scalar -> out
  dot_w3_kernel<<<(N + TB - 1) / TB, TB, 0, stream>>>(bufB, W3, tvec, N);
  out_init_kernel<<<(N + TB - 1) / TB, TB, 0, stream>>>(tvec, dinv, b3, out, N);
  scatter1_kernel<<<(E + TB - 1) / TB, TB, 0, stream>>>(src, dst, tvec, dinv, out, E);
}